// SoftAttentionDrop_64905545777531
// MI455X (gfx1250) — compile-verified
//
#include <hip/hip_runtime.h>

typedef __attribute__((ext_vector_type(2))) float v2f;
typedef __attribute__((ext_vector_type(8))) float v8f;

#define DIM        512
#define N_ROWS_TOT 32768
#define ROWS_PB    16
#define ITERS      128          // round(0.25 * 512)
#define EPSV       1e-12f
#define AS         516          // padded LDS row stride (516 % 64 == 4 -> conflict-free)

// Fused: logits = feature @ W^T via V_WMMA_F32_16X16X4_F32, then 128 sharpen
// iterations fully in registers using the invariant E = exp(-2*logit - rowmax),
// then out = feature * (1 - acc).
__global__ __launch_bounds__(256) void sad_fused_kernel(
    const float* __restrict__ feature,
    const float* __restrict__ Wm,
    float* __restrict__ out) {

    __shared__ float Afeat[ROWS_PB * AS];   // 33 KB: feature tile, reused at epilogue
    __shared__ float part[2][8][16];        // per-wave row partials
    __shared__ float rowtab[2][16];         // per-row rowmax / 1/rowsum

    const int tid  = threadIdx.x;
    const int lane = tid & 31;
    const int wid  = tid >> 5;              // 8 waves, wave w owns cols [64w, 64w+64)
    const int half = lane >> 4;
    const int l16  = lane & 15;
    const long r0  = (long)blockIdx.x * ROWS_PB;

    // ---- Stage 16 feature rows into LDS (coalesced float4) ----
    {
        const float4* src = (const float4*)(feature + r0 * DIM);
        #pragma unroll
        for (int i = 0; i < (ROWS_PB * (DIM / 4)) / 256; ++i) {
            int idx = tid + i * 256;
            int row = idx >> 7;             // DIM/4 == 128
            int c4  = idx & 127;
            float4 v = src[row * (DIM / 4) + c4];
            float* dst = &Afeat[row * AS + c4 * 4];
            dst[0] = v.x; dst[1] = v.y; dst[2] = v.z; dst[3] = v.w;
        }
    }
    __syncthreads();

    // ---- GEMM: 4 tiles of 16x16 per wave, K = 512 in steps of 4 ----
    v8f acc[4];
    #pragma unroll
    for (int t = 0; t < 4; ++t)
        #pragma unroll
        for (int e = 0; e < 8; ++e) acc[t][e] = 0.0f;

    const int colbase = wid * 64 + l16;
    #pragma unroll 4
    for (int k = 0; k < DIM; k += 4) {
        const int kk = k + 2 * half;        // A 16x4 f32 layout: lane=M, half/VGPR=K
        v2f a;
        const float* ap = &Afeat[l16 * AS + kk];
        a.x = ap[0];
        a.y = ap[1];
        #pragma unroll
        for (int t = 0; t < 4; ++t) {
            const float* bp = Wm + (size_t)(colbase + t * 16) * DIM + kk;  // B(k,n)=W[n][k]
            v2f b;
            b.x = bp[0];
            b.y = bp[1];
            acc[t] = __builtin_amdgcn_wmma_f32_16x16x4_f32(
                false, a, false, b, (short)0, acc[t], false, false);
        }
    }

    // ---- Row max of z = -2*logit over all 512 cols (for stable E) ----
    float p[8];
    #pragma unroll
    for (int v = 0; v < 8; ++v) {
        float m0 = fmaxf(fmaxf(-2.0f * acc[0][v], -2.0f * acc[1][v]),
                         fmaxf(-2.0f * acc[2][v], -2.0f * acc[3][v]));
        m0 = fmaxf(m0, __shfl_xor(m0, 1, 32));
        m0 = fmaxf(m0, __shfl_xor(m0, 2, 32));
        m0 = fmaxf(m0, __shfl_xor(m0, 4, 32));
        m0 = fmaxf(m0, __shfl_xor(m0, 8, 32));
        p[v] = m0;                           // all 16 lanes of the half hold it
    }
    if (l16 == 0) {
        #pragma unroll
        for (int v = 0; v < 8; ++v) part[0][wid][v + 8 * half] = p[v];
    }
    __syncthreads();
    if (tid < 16) {
        float m0 = part[0][0][tid];
        #pragma unroll
        for (int w = 1; w < 8; ++w) m0 = fmaxf(m0, part[0][w][tid]);
        rowtab[0][tid] = m0;
    }
    __syncthreads();

    // ---- E = exp(-2*logit - rowmax): iteration-invariant softmax numerator ----
    float E[4][8];
    #pragma unroll
    for (int v = 0; v < 8; ++v) {
        const float rmax = rowtab[0][v + 8 * half];
        #pragma unroll
        for (int t = 0; t < 4; ++t)
            E[t][v] = __expf(-2.0f * acc[t][v] - rmax);
    }

    // ---- 128 sharpen iterations: y_i = (m_i+eps)^2 E_i / sum; acc += y*m ----
    float accM[4][8];
    #pragma unroll
    for (int t = 0; t < 4; ++t)
        #pragma unroll
        for (int v = 0; v < 8; ++v) accM[t][v] = 0.0f;

    for (int it = 0; it < ITERS; ++it) {
        const int buf = it & 1;
        float wv[4][8];
        #pragma unroll
        for (int v = 0; v < 8; ++v) {
            float s = 0.0f;
            #pragma unroll
            for (int t = 0; t < 4; ++t) {
                float m  = 1.0f - accM[t][v];
                float mm = m + EPSV;
                float wl = mm * mm * E[t][v];
                wv[t][v] = wl;
                s += wl;
            }
            s += __shfl_xor(s, 1, 32);
            s += __shfl_xor(s, 2, 32);
            s += __shfl_xor(s, 4, 32);
            s += __shfl_xor(s, 8, 32);
            p[v] = s;
        }
        if (l16 == 0) {
            #pragma unroll
            for (int v = 0; v < 8; ++v) part[buf][wid][v + 8 * half] = p[v];
        }
        __syncthreads();
        if (tid < 16) {
            float s = 0.0f;
            #pragma unroll
            for (int w = 0; w < 8; ++w) s += part[buf][w][tid];
            rowtab[buf][tid] = __builtin_amdgcn_rcpf(s);
        }
        __syncthreads();
        #pragma unroll
        for (int v = 0; v < 8; ++v) {
            const float inv = rowtab[buf][v + 8 * half];
            #pragma unroll
            for (int t = 0; t < 4; ++t) {
                float m = 1.0f - accM[t][v];
                accM[t][v] = fmaf(wv[t][v] * inv, m, accM[t][v]);
            }
        }
    }

    // ---- out = feature * (1 - acc) ----
    #pragma unroll
    for (int v = 0; v < 8; ++v) {
        const int row = v + 8 * half;       // C/D layout: VGPR v, half -> M = v + 8*half
        #pragma unroll
        for (int t = 0; t < 4; ++t) {
            const int col = wid * 64 + t * 16 + l16;   // N = lane
            const float m = 1.0f - accM[t][v];
            out[(r0 + row) * DIM + col] = Afeat[row * AS + col] * m;
        }
    }
}

extern "C" void kernel_launch(void* const* d_in, const int* in_sizes, int n_in,
                              void* d_out, int out_size, void* d_ws, size_t ws_size,
                              hipStream_t stream) {
    (void)in_sizes; (void)n_in; (void)out_size; (void)d_ws; (void)ws_size;
    const float* feature = (const float*)d_in[0];
    const float* Wm      = (const float*)d_in[1];
    float* out           = (float*)d_out;
    dim3 grid(N_ROWS_TOT / ROWS_PB, 1, 1);
    sad_fused_kernel<<<grid, 256, 0, stream>>>(feature, Wm, out);
}